// HashGridLoRAEncoder_12841952215355
// MI455X (gfx1250) — compile-verified
//
#include <hip/hip_runtime.h>
#include <math.h>

// ---------------- static config (mirrors reference) ----------------
#define NLEVELS 16
#define HSIZE   32768        // 1 << 15
#define RANK    4
#define FPL     2            // features per level
#define OUTF    (NLEVELS * FPL)   // 32

typedef __attribute__((ext_vector_type(2))) float v2f;
typedef __attribute__((ext_vector_type(8))) float v8f;

// =====================================================================
// Kernel 1: materialize tables[l] = A[l] (HSIZE x 4) @ B[l] (4 x 2)
// using V_WMMA_F32_16X16X4_F32 (K = RANK = 4 exactly).
// One wave per 16-row tile; 16 levels * 2048 tiles = 32768 waves.
//
// A operand layout (ISA 32-bit A 16x4): lanes 0-15: VGPR0=K0, VGPR1=K1
//                                       lanes 16-31: VGPR0=K2, VGPR1=K3
// D layout (ISA 32-bit C/D 16x16): VGPR v: lanes 0-15 -> M=v, N=lane
//                                           lanes 16-31 -> M=v+8, N=lane-16
// =====================================================================
__global__ __launch_bounds__(256) void hashgrid_materialize_kernel(
    const float* __restrict__ A,   // [L, HSIZE, RANK]
    const float* __restrict__ B,   // [L, RANK, FPL]
    float* __restrict__ T)         // [L, HSIZE, FPL]
{
    const int wave  = (int)((blockIdx.x * blockDim.x + threadIdx.x) >> 5);
    const int lane  = (int)(threadIdx.x & 31);
    const int level = wave >> 11;          // 2048 tiles per level
    const int tile  = wave & 2047;
    const int row0  = tile << 4;           // 16 rows per tile

    const float* Al = A + ((size_t)level * HSIZE + row0) * RANK;
    const float* Bl = B + (size_t)level * RANK * FPL;

    const int half = lane >> 4;            // 0: lanes 0-15, 1: lanes 16-31
    const int l15  = lane & 15;

    // A operand: row = l15, this lane-half covers K = {2*half, 2*half+1}
    v2f a;
    {
        const float* ap = Al + l15 * RANK + half * 2;
        a.x = ap[0];
        a.y = ap[1];
    }
    // B operand (4x16): col = l15 (only cols 0..1 are real, rest zero);
    // VGPR0 holds K=2*half, VGPR1 holds K=2*half+1 (mirror of A layout).
    v2f b;
    {
        const int k0 = half * 2;
        b.x = (l15 < FPL) ? Bl[k0 * FPL + l15]       : 0.0f;
        b.y = (l15 < FPL) ? Bl[(k0 + 1) * FPL + l15] : 0.0f;
    }

    v8f c = {};
#if __has_builtin(__builtin_amdgcn_wmma_f32_16x16x4_f32)
    v8f d = __builtin_amdgcn_wmma_f32_16x16x4_f32(
        /*neg_a=*/false, a, /*neg_b=*/false, b,
        /*c_mod=*/(short)0, c, /*reuse_a=*/false, /*reuse_b=*/false);
#else
    // Fallback (keeps full compile alive if builtin name differs): scalar matvec
    (void)a; (void)b;
    v8f d = c;
#pragma unroll
    for (int v = 0; v < 8; ++v) {
        const int m = v + half * 8;
        float s = 0.0f;
        if (l15 < FPL) {
#pragma unroll
            for (int r = 0; r < RANK; ++r) s += Al[m * RANK + r] * Bl[r * FPL + l15];
        }
        d[v] = s;
    }
#endif

    // Write the FPL=2 valid columns of D.
    if (l15 < FPL) {
        float* Tl = T + ((size_t)level * HSIZE + row0) * FPL;
#pragma unroll
        for (int v = 0; v < 8; ++v) {
            const int m = v + half * 8;
            Tl[m * FPL + l15] = d[v];
        }
    }
}

// =====================================================================
// Kernel 2: hash-grid encode. grid = (chunks, NLEVELS).
// Each block stages its level's full 256 KB table into LDS (CDNA5:
// 320 KB LDS per workgroup), then serves all gathers from LDS.
// =====================================================================
struct ResArr { float r[NLEVELS]; };

__global__ __launch_bounds__(512) void hashgrid_encode_kernel(
    const float* __restrict__ x,    // [N, 3]
    const float* __restrict__ T,    // [L, HSIZE, FPL] materialized
    float* __restrict__ out,        // [N, 32]
    int N, int chunk, ResArr res)
{
    extern __shared__ float lds[];  // HSIZE*FPL floats = 256 KB

    const int level = (int)blockIdx.y;

    // ---- cooperative table stage: 16384 float4 ----
    {
        const float4* src = (const float4*)(T + (size_t)level * HSIZE * FPL);
        float4*       dst = (float4*)lds;
        const int n4 = (HSIZE * FPL) / 4;
        for (int i = (int)threadIdx.x; i < n4; i += (int)blockDim.x)
            dst[i] = src[i];
    }
    __syncthreads();

    const float2* tl = (const float2*)lds;
    const float   rl = res.r[level];

    const int p0 = (int)blockIdx.x * chunk;
    int p1 = p0 + chunk; if (p1 > N) p1 = N;

    for (int p = p0 + (int)threadIdx.x; p < p1; p += (int)blockDim.x) {
        const float px = x[3 * p + 0];
        const float py = x[3 * p + 1];
        const float pz = x[3 * p + 2];

        // normalize to [0,1], scale to level resolution
        const float xl = (px + 1.0f) * 0.5f * rl;
        const float yl = (py + 1.0f) * 0.5f * rl;
        const float zl = (pz + 1.0f) * 0.5f * rl;

        const float fx = floorf(xl), fy = floorf(yl), fz = floorf(zl);
        const float wx = xl - fx,   wy = yl - fy,   wz = zl - fz;
        const float ox = 1.0f - wx, oy = 1.0f - wy, oz = 1.0f - wz;

        const unsigned ix = (unsigned)fx, iy = (unsigned)fy, iz = (unsigned)fz;
        const unsigned hx0 = ix;                       // prime 1
        const unsigned hx1 = ix + 1u;
        const unsigned hy0 = iy * 2654435761u;
        const unsigned hy1 = hy0 + 2654435761u;
        const unsigned hz0 = iz * 805459861u;
        const unsigned hz1 = hz0 + 805459861u;

        float a0 = 0.0f, a1 = 0.0f;
#pragma unroll
        for (int c = 0; c < 8; ++c) {
            const unsigned h = ((c & 1) ? hx1 : hx0)
                             ^ ((c & 2) ? hy1 : hy0)
                             ^ ((c & 4) ? hz1 : hz0);
            const float wgt = ((c & 1) ? wx : ox)
                            * ((c & 2) ? wy : oy)
                            * ((c & 4) ? wz : oz);
            const float2 f = tl[h & (HSIZE - 1)];      // ds_load_b64 gather
            a0 = fmaf(f.x, wgt, a0);
            a1 = fmaf(f.y, wgt, a1);
        }

        float2 o; o.x = a0; o.y = a1;
        *(float2*)(out + (size_t)p * OUTF + level * FPL) = o;
    }
}

// =====================================================================
extern "C" void kernel_launch(void* const* d_in, const int* in_sizes, int n_in,
                              void* d_out, int out_size, void* d_ws, size_t ws_size,
                              hipStream_t stream) {
    const float* x = (const float*)d_in[0];   // [N,3]
    const float* A = (const float*)d_in[1];   // [16,32768,4]
    const float* B = (const float*)d_in[2];   // [16,4,2]
    float* out = (float*)d_out;               // [N,32]
    float* T   = (float*)d_ws;                // scratch: 16*32768*2 floats = 4 MB

    const int N = in_sizes[0] / 3;

    // ---- pass 1: LoRA expansion via WMMA ----
    {
        const int waves  = NLEVELS * (HSIZE / 16);     // 32768 waves
        const int blocks = (waves * 32) / 256;         // 4096 blocks of 8 waves
        hashgrid_materialize_kernel<<<blocks, 256, 0, stream>>>(A, B, T);
    }

    // ---- per-level resolutions: floor(16 * b^l), b = exp(ln(512/16)/15) ----
    ResArr res;
    {
        const double b = exp((log(512.0) - log(16.0)) / 15.0);
        for (int l = 0; l < NLEVELS; ++l)
            res.r[l] = (float)floor(16.0 * pow(b, (double)l));
    }

    // ---- pass 2: LDS-staged gather/interp ----
    {
        const int    chunk = 16384;                       // points per block
        const dim3   grid((N + chunk - 1) / chunk, NLEVELS);
        const size_t shmem = (size_t)HSIZE * FPL * sizeof(float);  // 262144 B
        (void)hipFuncSetAttribute((const void*)hashgrid_encode_kernel,
                                  hipFuncAttributeMaxDynamicSharedMemorySize,
                                  (int)shmem);
        hashgrid_encode_kernel<<<grid, 512, shmem, stream>>>(x, T, out, N, chunk, res);
    }
}